// SODE_76252849373555
// MI455X (gfx1250) — compile-verified
//
#include <hip/hip_runtime.h>

typedef __attribute__((ext_vector_type(2))) float v2f;
typedef __attribute__((ext_vector_type(8))) float v8f;

#define T_      64
#define DIM_    64
#define V_      25
#define NSTEP_  3
#define BN_     16384          // CLIPS * T
#define LEAK_   0.01f

#define P_      44             // state tile row pitch (dwords) -> conflict-free WMMA frag loads
#define PW_     68             // weight tile row pitch (dwords)
#define PA_     44             // A^T tile pitch
#define SLOTS_  50             // 1600 elems / 32 lanes
#define WPG_    4              // waves per workgroup (1 sample per wave)

// LDS floats: 3 weights (64*68) + AT (28*44) + 3 biases (64) + per-wave (2*64*44 + 64 pe)
#define SMEM_FLOATS_ (3*64*PW_ + 28*PA_ + 3*64 + WPG_*(2*64*P_ + 64))

#define O_W1_   0
#define O_W2_   (64*PW_)
#define O_WP_   (2*64*PW_)
#define O_AT_   (3*64*PW_)
#define O_B_    (O_AT_ + 28*PA_)
#define O_WAV_  (O_B_ + 3*64)

__device__ __forceinline__ float lrelu(float v) {
    // where(v>=0, v, LEAK*v) without cmp/cndmask: max(v,0) + LEAK*min(v,0)
    return __builtin_fmaf(LEAK_, __builtin_fminf(v, 0.0f), __builtin_fmaxf(v, 0.0f));
}

__global__ __launch_bounds__(WPG_*32, 1)
void sode_wmma_kernel(const float* __restrict__ fp,
                      const float* __restrict__ Ag,
                      const float* __restrict__ W1g, const float* __restrict__ b1g,
                      const float* __restrict__ W2g, const float* __restrict__ b2g,
                      const float* __restrict__ Wpg, const float* __restrict__ bpg,
                      const float* __restrict__ peg,
                      float* __restrict__ out)
{
    __shared__ float smem[SMEM_FLOATS_];

    const int tid = threadIdx.x;

    // ---- cooperative staging of constants into LDS (all via 32-bit LDS offsets) ----
    for (int i = tid; i < 64*64; i += WPG_*32) {
        int r = i >> 6, c = i & 63;
        smem[O_W1_ + r*PW_ + c] = W1g[i];
        smem[O_W2_ + r*PW_ + c] = W2g[i];
        smem[O_WP_ + r*PW_ + c] = Wpg[i];
    }
    for (int i = tid; i < 28*PA_; i += WPG_*32) smem[O_AT_ + i] = 0.0f; // zero K/N pad
    __syncthreads();
    for (int i = tid; i < V_*V_; i += WPG_*32) {
        int v = i / V_, u = i - v*V_;
        smem[O_AT_ + u*PA_ + v] = Ag[i];   // AT[u][v] = A[v][u]
    }
    for (int i = tid; i < 64; i += WPG_*32) {
        smem[O_B_ +       i] = b1g[i];
        smem[O_B_ +  64 + i] = b2g[i];
        smem[O_B_ + 128 + i] = bpg[i];
    }
    __syncthreads();

    // ---- per-wave setup ----
    const int wid  = tid >> 5;
    const int lane = tid & 31;
    const int n    = blockIdx.x * WPG_ + wid;      // sample id
    const int trow = n & (T_ - 1);                 // time index within clip
    const int half = lane >> 4;                    // 0 | 1
    const int mn   = lane & 15;                    // m or n within 16

    const int otA  = O_WAV_ + wid * (2*64*P_ + 64);
    const int otB  = otA + 64*P_;
    const int oPel = otB + 64*P_;

    // zero padding columns 25..31 of both state tiles (keeps garbage out of WMMA)
    for (int i = lane; i < 64*7; i += 32) {
        int r = i / 7, c = V_ + (i % 7);
        smem[otA + r*P_ + c] = 0.0f;
        smem[otB + r*P_ + c] = 0.0f;
    }

    // preload A^T B-fragments (loop invariant): k0 = 4*kbi + 2*half, rows >=25 are zero
    v2f atf[2][7];
#pragma unroll
    for (int noi = 0; noi < 2; ++noi)
#pragma unroll
        for (int kbi = 0; kbi < 7; ++kbi) {
            int k0 = kbi*4 + 2*half;
            atf[noi][kbi].x = smem[O_AT_ +  k0   *PA_ + noi*16 + mn];
            atf[noi][kbi].y = smem[O_AT_ + (k0+1)*PA_ + noi*16 + mn];
        }

    // ---- state in registers ----
    float y[SLOTS_], k1[SLOTS_], k2[SLOTS_], k3[SLOTS_];
    {
        const float* src  = fp  + (size_t)n * 1600;
        float*       dst0 = out + (size_t)n * 1600;      // step 0 = first_point
#pragma unroll
        for (int s = 0; s < SLOTS_; ++s) {
            y[s] = src[s*32 + lane];
            dst0[s*32 + lane] = y[s];
        }
    }

    // conv1x1 stage: out[o,v] = act( sum_c W[o,c] * in[c,v] + b[o] )
    // All B-fragments hoisted once; K-loop streams 2 A-fragments feeding
    // 4 independent WMMA accumulator chains (mo-pair x no-pair).
    auto conv = [&](int oW, int oBias, int oin, int oout, bool act) {
        v2f bf0[16], bf1[16];
#pragma unroll
        for (int kbi = 0; kbi < 16; ++kbi) {
            int k0 = kbi*4 + 2*half;
            bf0[kbi].x = smem[oin +  k0   *P_ +      mn];
            bf0[kbi].y = smem[oin + (k0+1)*P_ +      mn];
            bf1[kbi].x = smem[oin +  k0   *P_ + 16 + mn];
            bf1[kbi].y = smem[oin + (k0+1)*P_ + 16 + mn];
        }
#pragma unroll
        for (int mp = 0; mp < 2; ++mp) {
            const int mo0 = mp*32, mo1 = mp*32 + 16;
            v8f acc00, acc01, acc10, acc11;   // [mo][no]
#pragma unroll
            for (int r = 0; r < 8; ++r) {
                float bv0 = smem[oBias + mo0 + half*8 + r];
                float bv1 = smem[oBias + mo1 + half*8 + r];
                acc00[r] = bv0; acc01[r] = bv0;
                acc10[r] = bv1; acc11[r] = bv1;
            }
#pragma unroll
            for (int kbi = 0; kbi < 16; ++kbi) {
                int k0 = kbi*4 + 2*half;
                v2f a0, a1;
                a0.x = smem[oW + (mo0+mn)*PW_ + k0];
                a0.y = smem[oW + (mo0+mn)*PW_ + k0 + 1];
                a1.x = smem[oW + (mo1+mn)*PW_ + k0];
                a1.y = smem[oW + (mo1+mn)*PW_ + k0 + 1];
                acc00 = __builtin_amdgcn_wmma_f32_16x16x4_f32(
                            false, a0, false, bf0[kbi], (short)0, acc00, false, false);
                acc01 = __builtin_amdgcn_wmma_f32_16x16x4_f32(
                            false, a0, false, bf1[kbi], (short)0, acc01, false, false);
                acc10 = __builtin_amdgcn_wmma_f32_16x16x4_f32(
                            false, a1, false, bf0[kbi], (short)0, acc10, false, false);
                acc11 = __builtin_amdgcn_wmma_f32_16x16x4_f32(
                            false, a1, false, bf1[kbi], (short)0, acc11, false, false);
            }
#pragma unroll
            for (int r = 0; r < 8; ++r) {
                float v00 = acc00[r], v01 = acc01[r];
                float v10 = acc10[r], v11 = acc11[r];
                if (act) {
                    v00 = lrelu(v00); v01 = lrelu(v01);
                    v10 = lrelu(v10); v11 = lrelu(v11);
                }
                smem[oout + (mo0 + half*8 + r)*P_ +      mn] = v00;
                smem[oout + (mo0 + half*8 + r)*P_ + 16 + mn] = v01;
                smem[oout + (mo1 + half*8 + r)*P_ +      mn] = v10;
                smem[oout + (mo1 + half*8 + r)*P_ + 16 + mn] = v11;
            }
        }
    };

    // graph mixing: out[c,v] = sum_u in[c,u] * AT[u,v]   (K padded to 28, pad rows zero)
    // mo tiles paired + both noi -> 4 independent WMMA chains; B operands in registers.
    auto amix = [&](int oin, int oout) {
#pragma unroll
        for (int mp = 0; mp < 2; ++mp) {
            const int mo0 = mp*32, mo1 = mp*32 + 16;
            v8f acc00, acc01, acc10, acc11;   // [mo][noi]
#pragma unroll
            for (int r = 0; r < 8; ++r) {
                acc00[r] = 0.0f; acc01[r] = 0.0f;
                acc10[r] = 0.0f; acc11[r] = 0.0f;
            }
#pragma unroll
            for (int kbi = 0; kbi < 7; ++kbi) {
                int k0 = kbi*4 + 2*half;
                v2f a0, a1;
                a0.x = smem[oin + (mo0+mn)*P_ + k0];
                a0.y = smem[oin + (mo0+mn)*P_ + k0 + 1];
                a1.x = smem[oin + (mo1+mn)*P_ + k0];
                a1.y = smem[oin + (mo1+mn)*P_ + k0 + 1];
                acc00 = __builtin_amdgcn_wmma_f32_16x16x4_f32(
                            false, a0, false, atf[0][kbi], (short)0, acc00, false, false);
                acc01 = __builtin_amdgcn_wmma_f32_16x16x4_f32(
                            false, a0, false, atf[1][kbi], (short)0, acc01, false, false);
                acc10 = __builtin_amdgcn_wmma_f32_16x16x4_f32(
                            false, a1, false, atf[0][kbi], (short)0, acc10, false, false);
                acc11 = __builtin_amdgcn_wmma_f32_16x16x4_f32(
                            false, a1, false, atf[1][kbi], (short)0, acc11, false, false);
            }
#pragma unroll
            for (int r = 0; r < 8; ++r) {
                smem[oout + (mo0 + half*8 + r)*P_ +      mn] = acc00[r];
                smem[oout + (mo0 + half*8 + r)*P_ + 16 + mn] = acc01[r];
                smem[oout + (mo1 + half*8 + r)*P_ +      mn] = acc10[r];
                smem[oout + (mo1 + half*8 + r)*P_ + 16 + mn] = acc11[r];
            }
        }
    };

    // build ode input u (+ PE row) into tA from register state
    auto compose = [&](int mode, int ti) {
        const float* pg = peg + (size_t)(ti + trow) * DIM_;
        smem[oPel + lane]      = pg[lane];
        smem[oPel + lane + 32] = pg[lane + 32];
#pragma unroll
        for (int s = 0; s < SLOTS_; ++s) {
            int f = s*32 + lane;
            int c = f / V_;
            int v = f - c * V_;
            float u;
            if      (mode == 0) u = y[s];
            else if (mode == 1) u = y[s] + k1[s] * (1.0f/3.0f);
            else if (mode == 2) u = y[s] + (k2[s] - k1[s] * (1.0f/3.0f));
            else                u = y[s] + (k1[s] - k2[s] + k3[s]);
            smem[otA + c*P_ + v] = u + smem[oPel + c];
        }
    };

    // one ode_func evaluation: result lands in tile B
    auto ode = [&](int mode, int ti) {
        compose(mode, ti);
        amix(otA, otB);
        conv(O_W1_, O_B_,       otB, otA, true);
        amix(otA, otB);
        conv(O_W2_, O_B_ + 64,  otB, otA, true);
        conv(O_WP_, O_B_ + 128, otA, otB, false);
    };

    auto readback = [&](float* dst) {
#pragma unroll
        for (int s = 0; s < SLOTS_; ++s) {
            int f = s*32 + lane;
            int c = f / V_;
            int v = f - c * V_;
            dst[s] = smem[otB + c*P_ + v];
        }
    };

    // ---- RK4 scan: t0 = st, dt = 1 (time_steps == arange) ----
    for (int st = 0; st < NSTEP_; ++st) {
        ode(0, st);     readback(k1);
        ode(1, st);     readback(k2);
        ode(2, st);     readback(k3);
        ode(3, st + 1); // k4 in tile B, consumed in-place
#pragma unroll
        for (int s = 0; s < SLOTS_; ++s) {
            int f = s*32 + lane;
            int c = f / V_;
            int v = f - c * V_;
            float k4 = smem[otB + c*P_ + v];
            y[s] += (k1[s] + 3.0f*(k2[s] + k3[s]) + k4) * 0.125f;
        }
        float* dsto = out + ((size_t)(st+1) * BN_ + n) * 1600;
#pragma unroll
        for (int s = 0; s < SLOTS_; ++s)
            dsto[s*32 + lane] = y[s];
    }
}

extern "C" void kernel_launch(void* const* d_in, const int* in_sizes, int n_in,
                              void* d_out, int out_size, void* d_ws, size_t ws_size,
                              hipStream_t stream) {
    (void)in_sizes; (void)n_in; (void)out_size; (void)d_ws; (void)ws_size;
    const float* fp = (const float*)d_in[0];
    // d_in[1] = time_steps (== arange(4), folded into constants)
    const float* A  = (const float*)d_in[2];
    const float* W1 = (const float*)d_in[3];
    const float* b1 = (const float*)d_in[4];
    const float* W2 = (const float*)d_in[5];
    const float* b2 = (const float*)d_in[6];
    const float* Wp = (const float*)d_in[7];
    const float* bp = (const float*)d_in[8];
    const float* pe = (const float*)d_in[9];
    float* out = (float*)d_out;

    dim3 grid(BN_ / WPG_);
    dim3 block(WPG_ * 32);
    hipLaunchKernelGGL(sode_wmma_kernel, grid, block, 0, stream,
                       fp, A, W1, b1, W2, b2, Wp, bp, pe, out);
}